// GaussianSpot_87763361726799
// MI455X (gfx1250) — compile-verified
//
#include <hip/hip_runtime.h>

typedef __attribute__((ext_vector_type(2))) float v2f;
typedef __attribute__((ext_vector_type(8))) float v8f;

#define DD   14
#define FF   400
#define NPIX 196              // 14*14
#define WPB  8                // waves per block (wave32)
#define BF_TOTAL (256 * 400)  // B*F = 102400

// One wave32 renders one (b,f) 14x14 Gaussian-spot tile with a single
// V_WMMA_F32_16X16X4_F32.  Separable PSF:  O = Gx(16x4) * Gy^T(4x16), with
// coefA folded into Gx, and the background folded into the spare K=2 column
// (A[:,2]=bg, B[2,:]=1) so the accumulator C is the inline-0 operand.
__global__ __launch_bounds__(256) void gspot_wmma_kernel(
    const int*    __restrict__ batch_idx,   // (B,)
    const uchar2* __restrict__ m_mask,      // (B,F,K=2) bool, paired
    const float2* __restrict__ height,      // (B,F,2)
    const float2* __restrict__ width,       // (B,F,2)
    const float2* __restrict__ x0v,         // (B,F,2)
    const float2* __restrict__ y0v,         // (B,F,2)
    const float*  __restrict__ background,  // (B,F)
    const float2* __restrict__ target_locs, // (N,F,1,1,1,2) -> (N*F) float2
    float*        __restrict__ out)         // (B,F,14,14)
{
    __shared__ float tile[WPB * 256];

    const int tid  = threadIdx.x;
    const int wave = tid >> 5;
    const int lane = tid & 31;
    // Wave-uniform tile id: make uniformity explicit so all parameter loads
    // become scalar (SMEM) loads and address math goes to the SALU pipe.
    const int bf = __builtin_amdgcn_readfirstlane(blockIdx.x * WPB + wave);

    const int b = bf / FF;
    const int f = bf - b * FF;

    // ---- wave-uniform spot parameters (scalar loads), vectorized over K=2 ----
    const int    n   = batch_idx[b];
    const float2 tl  = target_locs[n * FF + f];
    const float  bg  = background[bf];
    const uchar2 mk  = m_mask[bf];
    const float2 hh  = height[bf];
    const float2 ww  = width[bf];
    const float2 xx  = x0v[bf];
    const float2 yy  = y0v[bf];

    const float INV2PI = 0.15915494309189535f;  // 1/(2*pi)
    const float h0  = mk.x ? hh.x : 0.0f;
    const float h1  = mk.y ? hh.y : 0.0f;
    const float iw0 = 1.0f / (ww.x * ww.x);     // 1/w^2
    const float iw1 = 1.0f / (ww.y * ww.y);
    const float cA0 = h0 * INV2PI * iw0;        // h / (2*pi*w^2)
    const float cA1 = h1 * INV2PI * iw1;
    const float cB0 = -0.5f * iw0;
    const float cB1 = -0.5f * iw1;
    const float cx0 = tl.x + xx.x, cx1 = tl.x + xx.y;
    const float cy0 = tl.y + yy.x, cy1 = tl.y + yy.y;

    // ---- WMMA operands (ISA 7.12.2, wave32) ----
    // A (16x4): lanes 0-15 hold K=0/1 for row M=lane; lanes 16-31 hold K=2/3.
    // B (4x16): lanes 0-15 hold K=0/1 at col N=lane; lanes 16-31 hold K=2/3.
    // Spare column K=2 carries the background: A[m][2]=bg, B[2][n]=1.
    const float p = (float)(lane & 15);
    float a0, a1, b0, b1;
    if (lane < 16) {
        const float dx0 = p - cx0, dx1 = p - cx1;
        const float dy0 = p - cy0, dy1 = p - cy1;
        a0 = cA0 * __expf(cB0 * dx0 * dx0);
        a1 = cA1 * __expf(cB1 * dx1 * dx1);
        b0 = __expf(cB0 * dy0 * dy0);
        b1 = __expf(cB1 * dy1 * dy1);
    } else {
        a0 = bg;    // A[m][2]
        a1 = 0.0f;  // A[m][3]
        b0 = 1.0f;  // B[2][n]
        b1 = 0.0f;  // B[3][n]
    }
    v2f Am; Am.x = a0; Am.y = a1;
    v2f Bm; Bm.x = b0; Bm.y = b1;
    v8f Cm = {};  // inline-0 accumulator

    // D = A*B + 0   (neg_a, A, neg_b, B, c_mod, C, reuse_a, reuse_b)
    v8f Dm = __builtin_amdgcn_wmma_f32_16x16x4_f32(
        false, Am, false, Bm, (short)0, Cm, false, false);

    // ---- scatter D into final image layout in LDS (immediate DS offsets) ----
    // D layout: lanes 0-15 hold rows M=0..7 at col N=lane; lanes 16-31 rows 8..15.
    // wtile[M*14 + N] = D[M][N]; cols 14,15 masked; rows 14,15 land in the unused
    // tail (words 196..223) of the 256-float tile -- never read back.
    float* wtile = &tile[wave * 256];
    const int half = lane >> 4;
    const int col  = lane & 15;
    if (col < DD) {
        float* base = wtile + half * (8 * DD) + col;
#pragma unroll
        for (int r = 0; r < 8; ++r) base[r * DD] = Dm[r];  // ds_store offset:r*56
    }

    // intra-wave producer/consumer handoff (DS ops of one wave are in-order)
    asm volatile("s_wait_dscnt 0" ::: "memory");

    // ---- linear, fully-coalesced b128 copy: LDS -> global (SADDR form) ----
    float* outImg = out + (size_t)bf * NPIX;
#pragma unroll
    for (int it = 0; it < 2; ++it) {
        const int j = lane + it * 32;
        if (j < NPIX / 4) {                                 // 49 float4 chunks
            const float4 v = *(const float4*)(wtile + 4 * j);
            *(float4*)(outImg + 4 * j) = v;
        }
    }
}

extern "C" void kernel_launch(void* const* d_in, const int* in_sizes, int n_in,
                              void* d_out, int out_size, void* d_ws, size_t ws_size,
                              hipStream_t stream) {
    (void)in_sizes; (void)n_in; (void)out_size; (void)d_ws; (void)ws_size;
    const int*    batch_idx   = (const int*)d_in[0];
    const uchar2* m_mask      = (const uchar2*)d_in[1];   // jax bool, K=2 pairs
    const float2* height      = (const float2*)d_in[2];
    const float2* width       = (const float2*)d_in[3];
    const float2* x0v         = (const float2*)d_in[4];
    const float2* y0v         = (const float2*)d_in[5];
    const float*  background  = (const float*)d_in[6];
    const float2* target_locs = (const float2*)d_in[7];
    // d_in[8] = pixel_pos: implicit in the grid, unused
    float* out = (float*)d_out;

    dim3 grid(BF_TOTAL / WPB);  // 12800 blocks x 256 threads (8 waves each)
    gspot_wmma_kernel<<<grid, dim3(256), 0, stream>>>(
        batch_idx, m_mask, height, width, x0v, y0v, background, target_locs, out);
}